// SemiCurrSinkhornKnopp_15857019256982
// MI455X (gfx1250) — compile-verified
//
#include <hip/hip_runtime.h>
#include <math.h>

// Problem constants (match reference)
#define KCOLS   512
#define KP1     513
#define RHO_    0.5f
#define STOPERR_ 1e-6f
#define NITER   1000

// Row-pass config: 8 waves/block, one 16-row tile per wave.
#define WPB 8
#define TPB (WPB * 32)

typedef float v2f __attribute__((ext_vector_type(2)));
typedef float v8f __attribute__((ext_vector_type(8)));

// D = A(16x4 f32) * B(4x16 f32) + C(16x16 f32), wave32 striped.
__device__ __forceinline__ v8f wmma4(v2f a, v2f b, v8f c) {
  return __builtin_amdgcn_wmma_f32_16x16x4_f32(false, a, false, b, (short)0, c, false, false);
}

// ---------------------------------------------------------------------------
// Kernel 1: per-row logsumexp + Q = exp(10*(x - lse)) cached into d_out.
// One block (128 threads) per row of 512.
// ---------------------------------------------------------------------------
__global__ __launch_bounds__(128) void lse_q_kernel(const float* __restrict__ logits,
                                                    float* __restrict__ Q,
                                                    float* __restrict__ lse) {
  __shared__ float red[128];
  const int row = blockIdx.x;
  const int tid = threadIdx.x;
  const float* x = logits + (size_t)row * KCOLS;

  float m = -3.402823466e38f;
  for (int j = tid; j < KCOLS; j += 128) m = fmaxf(m, x[j]);
  red[tid] = m;
  __syncthreads();
  for (int s = 64; s > 0; s >>= 1) {
    if (tid < s) red[tid] = fmaxf(red[tid], red[tid + s]);
    __syncthreads();
  }
  const float M = red[0];
  __syncthreads();

  float sum = 0.f;
  for (int j = tid; j < KCOLS; j += 128) sum += expf(x[j] - M);
  red[tid] = sum;
  __syncthreads();
  for (int s = 64; s > 0; s >>= 1) {
    if (tid < s) red[tid] += red[tid + s];
    __syncthreads();
  }
  const float L = M + logf(red[0]);
  if (tid == 0) lse[row] = L;

  float* qr = Q + (size_t)row * KCOLS;
  for (int j = tid; j < KCOLS; j += 128) qr[j] = expf(10.f * (x[j] - L));
}

// ---------------------------------------------------------------------------
// Kernel 2: init b = 1/(k+1), ct = 0, flag = 0, ticket = 0.
// ---------------------------------------------------------------------------
__global__ void init_kernel(float* __restrict__ b, float* __restrict__ ct,
                            int* __restrict__ flag, int* __restrict__ ticket) {
  const int j = blockIdx.x * blockDim.x + threadIdx.x;
  if (j < KP1) {
    b[j] = 1.0f / (float)KP1;
    ct[j] = 0.f;
  }
  if (j == 0) { *flag = 0; *ticket = 0; }
}

// ---------------------------------------------------------------------------
// Kernel 3 (per iteration): fused Sinkhorn row pass + ticketed b-update.
//   dir1: d_i = (Q b)_i via WMMA (b in column N=0 of B)   -> a_i = Pa/(d_i+b_k)
//   dir2: ct_j += sum_i Q[i,j] a_i via WMMA (a in row M=0 of A)
//   last block: b_j = (Pb_j/ct_j)^(fi for j<k), err = ||db||, flag, ct=0.
// ---------------------------------------------------------------------------
__global__ __launch_bounds__(TPB) void rowpass_kernel(const float* __restrict__ Q,
                                                      float* __restrict__ bvec,
                                                      float* __restrict__ ct,
                                                      float* __restrict__ aout,
                                                      int* __restrict__ flag,
                                                      int* __restrict__ ticket,
                                                      float inv_n) {
  if (*flag != 0) return;  // converged earlier: whole dispatch is a no-op

  __shared__ float bsh[KP1 + 1];  // +1 pad so v2f loads at j=512 stay in-bounds
  __shared__ float ctsh[KP1];
  __shared__ float ash[WPB][16];
  __shared__ float red[TPB];
  __shared__ int winner;

  const int tid = threadIdx.x;
  for (int j = tid; j < KP1; j += TPB) {
    bsh[j] = bvec[j];
    ctsh[j] = 0.f;
  }
  if (tid == 0) bsh[KP1] = 0.f;
  __syncthreads();

  const int wv = tid >> 5;
  const int lane = tid & 31;
  const int mrow = lane & 15;   // M index within tile / N index in dir2
  const int khalf = lane >> 4;  // 0: K{0,1}, 1: K{2,3}
  const bool head = (mrow == 0);
  const int row0 = (blockIdx.x * WPB + wv) * 16;

  // ---- direction 1: 16 row-dots d = Q_tile @ b -------------------------
  // A 16x4: lane<16 -> (M=lane, K0/K1); lane>=16 -> (M, K2/K3)
  // B 4x16: only column N=0 is non-zero and carries the b chunk.
  const float* qrow = Q + (size_t)(row0 + mrow) * KCOLS;
  v8f C = {0.f, 0.f, 0.f, 0.f, 0.f, 0.f, 0.f, 0.f};
  for (int k0 = 0; k0 < KCOLS; k0 += 4) {
    v2f A = *(const v2f*)(qrow + k0 + 2 * khalf);            // global_load_b64
    v2f bb = *(const v2f*)(&bsh[k0 + 2 * khalf]);            // ds_load_b64, all lanes
    v2f B;
    B.x = head ? bb.x : 0.f;                                 // v_cndmask (no exec dance)
    B.y = head ? bb.y : 0.f;
    C = wmma4(A, B, C);
  }

  // a_i = Pa / (d_i + b_dummy);  C col N=0: lane0 holds M=0..7, lane16 M=8..15
  const float bdum = bsh[KCOLS];
  if (head) {
    float s = 0.f;
#pragma unroll
    for (int v = 0; v < 8; ++v) {
      const float av = inv_n / (C[v] + bdum);
      ash[wv][khalf * 8 + v] = av;
      aout[row0 + khalf * 8 + v] = av;
      s += av;
    }
    atomicAdd(&ctsh[KCOLS], s);  // dummy column: sum(a)
  }
  __syncthreads();

  // ---- direction 2: ct_j += a^T * Q_tile --------------------------------
  // A 16x4: only row M=0 is non-zero and carries a[r..r+3].
  // Hoist the 4 A fragments (r = 0,4,8,12): they do not depend on c0.
  v2f A2f[4];
#pragma unroll
  for (int ri = 0; ri < 4; ++ri) {
    v2f aa = *(const v2f*)(&ash[wv][ri * 4 + 2 * khalf]);    // ds_load_b64, all lanes
    A2f[ri].x = head ? aa.x : 0.f;
    A2f[ri].y = head ? aa.y : 0.f;
  }

  for (int c0 = 0; c0 < KCOLS; c0 += 16) {
    v8f C2 = {0.f, 0.f, 0.f, 0.f, 0.f, 0.f, 0.f, 0.f};
    const float* qb = Q + (size_t)row0 * KCOLS + c0 + mrow;
#pragma unroll
    for (int ri = 0; ri < 4; ++ri) {
      const int r = ri * 4;
      v2f B2;  // B 4x16 = Q rows r..r+3, cols c0..c0+15
      B2.x = qb[(size_t)(r + 2 * khalf) * KCOLS];
      B2.y = qb[(size_t)(r + 2 * khalf + 1) * KCOLS];
      C2 = wmma4(A2f[ri], B2, C2);
    }
    if (lane < 16) atomicAdd(&ctsh[c0 + lane], C2[0]);  // row M=0, N=lane
  }
  __syncthreads();

  for (int j = tid; j < KP1; j += TPB) atomicAdd(&ct[j], ctsh[j]);

  // ---- ticket: last block to finish performs the b-update ---------------
  if (tid == 0) {
    __threadfence();
    const int old = atomicAdd(ticket, 1);
    winner = (old == (int)gridDim.x - 1) ? 1 : 0;
  }
  __syncthreads();
  if (!winner) return;

  __threadfence();  // acquire: make all blocks' ct atomics visible
  const float fi = 1.0f / 1.1f;           // GAMMA/(GAMMA+EPSILON)
  const float pbk = RHO_ / (float)KCOLS;  // Pb for real columns
  float local = 0.f;
  for (int j = tid; j < KP1; j += TPB) {
    const float pb = (j < KCOLS) ? pbk : (1.0f - RHO_);
    float bn = pb / ct[j];
    if (j < KCOLS) bn = powf(bn, fi);  // semi-relaxed KL step (not on dummy)
    const float d = bn - bsh[j];       // bsh still holds previous b
    local += d * d;
    bvec[j] = bn;
    ct[j] = 0.f;  // ready for next iteration
  }
  red[tid] = local;
  __syncthreads();
  for (int s = TPB / 2; s > 0; s >>= 1) {
    if (tid < s) red[tid] += red[tid + s];
    __syncthreads();
  }
  if (tid == 0) {
    if (sqrtf(red[0]) <= STOPERR_) *flag = 1;
    *ticket = 0;
  }
}

// ---------------------------------------------------------------------------
// Kernel 4: plan = n * a_i * Q_ij * b_j, in place over d_out (Q cache).
// ---------------------------------------------------------------------------
__global__ __launch_bounds__(256) void plan_kernel(float* __restrict__ Q,
                                                   const float* __restrict__ a,
                                                   const float* __restrict__ bvec,
                                                   float nscale) {
  const size_t idx = ((size_t)blockIdx.x * 256 + threadIdx.x) * 4;
  const int row = (int)(idx >> 9);   // /512
  const int col = (int)(idx & 511);
  float4 q = *reinterpret_cast<float4*>(Q + idx);
  const float s = a[row] * nscale;
  q.x *= s * bvec[col + 0];
  q.y *= s * bvec[col + 1];
  q.z *= s * bvec[col + 2];
  q.w *= s * bvec[col + 3];
  *reinterpret_cast<float4*>(Q + idx) = q;
}

// ---------------------------------------------------------------------------
extern "C" void kernel_launch(void* const* d_in, const int* in_sizes, int n_in,
                              void* d_out, int out_size, void* d_ws, size_t ws_size,
                              hipStream_t stream) {
  const float* logits = (const float*)d_in[0];
  const int n = in_sizes[0] / KCOLS;  // 131072

  float* Q = (float*)d_out;  // Q cache lives in d_out; overwritten by plan at the end
  float* wsf = (float*)d_ws;
  float* bvec = wsf;                 // [513]
  float* ct   = wsf + 513;           // [513]
  int* flag   = (int*)(wsf + 1026);  // [1]
  int* ticket = (int*)(wsf + 1027);  // [1]
  float* aog  = wsf + 1032;          // [n]
  float* lse  = aog + n;             // [n]

  lse_q_kernel<<<n, 128, 0, stream>>>(logits, Q, lse);
  init_kernel<<<(KP1 + 255) / 256, 256, 0, stream>>>(bvec, ct, flag, ticket);

  const int blocks = n / (16 * WPB);  // 1024
  const float inv_n = 1.0f / (float)n;
  for (int it = 0; it < NITER; ++it) {
    rowpass_kernel<<<blocks, TPB, 0, stream>>>(Q, bvec, ct, aog, flag, ticket, inv_n);
  }

  const size_t vec4 = (size_t)n * KCOLS / 4;
  plan_kernel<<<(unsigned)(vec4 / 256), 256, 0, stream>>>(Q, aog, bvec, (float)n);
}